// GraphTransformer_layer_31928786879184
// MI455X (gfx1250) — compile-verified
//
#include <hip/hip_runtime.h>

#define N_NODES 10000
#define DEG     64
#define DH      128      // hidden dim
#define LN_EPS  1e-5f
#define M_PAD   10016    // 626 * 16

typedef __attribute__((ext_vector_type(2))) float v2f;
typedef __attribute__((ext_vector_type(8))) float v8f;

// ---------------- LayerNorm: one wave (32 lanes) per row, float4 per lane ----
__global__ __launch_bounds__(256) void ln_kernel(const float* __restrict__ x,
                                                 const float* __restrict__ g,
                                                 const float* __restrict__ b,
                                                 float* __restrict__ out,
                                                 int nvalid, int ntotal) {
  int row  = (int)((blockIdx.x * blockDim.x + threadIdx.x) >> 5);
  int lane = threadIdx.x & 31;
  if (row >= ntotal) return;
  float* orow = out + (size_t)row * DH;
  if (row >= nvalid) {                       // zero-fill pad rows (no OOB read)
    float4 z = make_float4(0.f, 0.f, 0.f, 0.f);
    *(float4*)(orow + lane * 4) = z;
    return;
  }
  const float* xrow = x + (size_t)row * DH;
  float4 v = *(const float4*)(xrow + lane * 4);
  float s = v.x + v.y + v.z + v.w;
  for (int off = 16; off >= 1; off >>= 1) s += __shfl_xor(s, off, 32);
  float mu = s * (1.0f / DH);
  float dx0 = v.x - mu, dx1 = v.y - mu, dx2 = v.z - mu, dx3 = v.w - mu;
  float q = dx0 * dx0 + dx1 * dx1 + dx2 * dx2 + dx3 * dx3;
  for (int off = 16; off >= 1; off >>= 1) q += __shfl_xor(q, off, 32);
  float rs = rsqrtf(q * (1.0f / DH) + LN_EPS);
  float4 gg = *(const float4*)(g + lane * 4);
  float4 bb = *(const float4*)(b + lane * 4);
  float4 o;
  o.x = dx0 * rs * gg.x + bb.x;
  o.y = dx1 * rs * gg.y + bb.y;
  o.z = dx2 * rs * gg.z + bb.z;
  o.w = dx3 * rs * gg.w + bb.w;
  *(float4*)(orow + lane * 4) = o;
}

// ---------------- GEMM: out[M,128] = act(A[M,128] @ W[128,128] + bias) (+res)
// One wave -> 16-row x 128-col strip via 8x V_WMMA_F32_16X16X4_F32 per k-step.
__global__ __launch_bounds__(128) void gemm128_wmma(const float* __restrict__ A,
                                                    const float* __restrict__ W,
                                                    const float* __restrict__ bias,
                                                    const float* __restrict__ res,
                                                    float* __restrict__ out,
                                                    int Mtot, int Mvalid, int relu) {
  int lane  = threadIdx.x & 31;
  int strip = blockIdx.x * 4 + (threadIdx.x >> 5);   // wave-uniform
  int m0 = strip * 16;
  if (m0 >= Mtot) return;                            // whole-wave exit, EXEC stays full
  int lr = lane & 15;     // A: row within tile; B: col within tile
  int hi = lane >> 4;     // 0: K pair {0,1}; 1: K pair {2,3}

  const float* Arow = A + (size_t)(m0 + lr) * DH;

  v8f acc[8];
  v8f zero = {0.f, 0.f, 0.f, 0.f, 0.f, 0.f, 0.f, 0.f};
#pragma unroll
  for (int i = 0; i < 8; ++i) acc[i] = zero;

  for (int k = 0; k < DH; k += 4) {
    float2 af = *(const float2*)(Arow + k + hi * 2);     // A[m, k+hi*2 .. +1]
    v2f a; a.x = af.x; a.y = af.y;
    const float* Wr = W + (size_t)(k + hi * 2) * DH;     // rows k+hi*2, k+hi*2+1
#pragma unroll
    for (int nt = 0; nt < 8; ++nt) {
      v2f bf;
      bf.x = Wr[nt * 16 + lr];
      bf.y = Wr[DH + nt * 16 + lr];
      acc[nt] = __builtin_amdgcn_wmma_f32_16x16x4_f32(
          /*neg_a=*/false, a, /*neg_b=*/false, bf,
          /*c_mod=*/(short)0, acc[nt], /*reuse_a=*/false, /*reuse_b=*/false);
    }
  }

#pragma unroll
  for (int nt = 0; nt < 8; ++nt) {
#pragma unroll
    for (int r = 0; r < 8; ++r) {
      int row = m0 + r + hi * 8;                 // C/D layout: vgpr r -> M=r / r+8
      int col = nt * 16 + lr;
      if (row < Mvalid) {
        float vv = acc[nt][r] + bias[col];
        if (relu) vv = fmaxf(vv, 0.f);
        if (res)  vv += res[(size_t)row * DH + col];
        out[(size_t)row * DH + col] = vv;
      }
    }
  }
}

// ---------------- Edge attention: block(128) per dst node; t = head*16 + dim -
__global__ __launch_bounds__(128) void attn_kernel(const float* __restrict__ Q,
                                                   const float* __restrict__ Km,
                                                   const float* __restrict__ Vm,
                                                   const int* __restrict__ src,
                                                   float* __restrict__ gmha,
                                                   int nvalid, int ntotal) {
  int n = blockIdx.x;
  int t = threadIdx.x;                    // 0..127 ; head = t>>4, dim = t&15
  if (n >= ntotal) return;
  if (n >= nvalid) { gmha[(size_t)n * DH + t] = 0.f; return; }   // zero pad rows
  float q = Q[(size_t)n * DH + t];
  const int* es = src + (size_t)n * DEG;  // dst = repeat(arange(N), DEG)
  float accn = 0.f, accd = 0.f;
  for (int e = 0; e < DEG; ++e) {
    int s = es[e];
    float p = Km[(size_t)s * DH + t] * q;
    // reduce over the 16 dims of this head (16-lane groups align with heads)
    p += __shfl_xor(p, 8, 16);
    p += __shfl_xor(p, 4, 16);
    p += __shfl_xor(p, 2, 16);
    p += __shfl_xor(p, 1, 16);
    float ex = __expf(p * 0.25f);         // / sqrt(HD=16)
    accn += ex * Vm[(size_t)s * DH + t];
    accd += ex;
  }
  gmha[(size_t)n * DH + t] = accn / accd;
}

extern "C" void kernel_launch(void* const* d_in, const int* in_sizes, int n_in,
                              void* d_out, int out_size, void* d_ws, size_t ws_size,
                              hipStream_t stream) {
  const float* h     = (const float*)d_in[0];
  const int*   src   = (const int*)  d_in[1];
  // d_in[2] = dst (implicit: repeat(arange(N), DEG)) — unused
  const float* WQ_w  = (const float*)d_in[3];
  const float* WQ_b  = (const float*)d_in[4];
  const float* WK_w  = (const float*)d_in[5];
  const float* WK_b  = (const float*)d_in[6];
  const float* WV_w  = (const float*)d_in[7];
  const float* WV_b  = (const float*)d_in[8];
  const float* WO_w  = (const float*)d_in[9];
  const float* WO_b  = (const float*)d_in[10];
  const float* ln1_g = (const float*)d_in[11];
  const float* ln1_b = (const float*)d_in[12];
  const float* ln2_g = (const float*)d_in[13];
  const float* ln2_b = (const float*)d_in[14];
  const float* W1_w  = (const float*)d_in[15];
  const float* W1_b  = (const float*)d_in[16];
  const float* W2_w  = (const float*)d_in[17];
  const float* W2_b  = (const float*)d_in[18];
  float* out = (float*)d_out;

  // Workspace: 6 padded [M_PAD,128] fp32 buffers (~31 MB), with reuse.
  const size_t BUF = (size_t)M_PAD * DH;
  float* ws  = (float*)d_ws;
  float* hn  = ws + 0 * BUF;   // LN1 out
  float* Qb  = ws + 1 * BUF;   // Q
  float* Kb  = ws + 2 * BUF;   // K; later reused as hn2 (LN2 out)
  float* Vb  = ws + 3 * BUF;   // V; later reused as MLP hidden
  float* Qq  = ws + 4 * BUF;   // extra slot so gMHA doesn't alias live Q
  float* h2  = ws + 5 * BUF;

  const int lnGrid   = M_PAD / 8;                 // 8 rows per 256-thread block
  const int gemmGrid = (M_PAD / 16 + 3) / 4;      // 4 strips (waves) per block

  // 1) hn = LN1(h)
  ln_kernel<<<lnGrid, 256, 0, stream>>>(h, ln1_g, ln1_b, hn, N_NODES, M_PAD);
  // 2) Q,K,V projections (WMMA f32)
  gemm128_wmma<<<gemmGrid, 128, 0, stream>>>(hn, WQ_w, WQ_b, nullptr, Qb, M_PAD, M_PAD, 0);
  gemm128_wmma<<<gemmGrid, 128, 0, stream>>>(hn, WK_w, WK_b, nullptr, Kb, M_PAD, M_PAD, 0);
  gemm128_wmma<<<gemmGrid, 128, 0, stream>>>(hn, WV_w, WV_b, nullptr, Vb, M_PAD, M_PAD, 0);
  // 3) edge attention -> gMHA (reuse hn's slot? keep separate slot Qq for clarity)
  attn_kernel<<<M_PAD, 128, 0, stream>>>(Qb, Kb, Vb, src, Qq, N_NODES, M_PAD);
  // 4) h2 = h + gMHA @ WO + bO   (residual read/store guarded to N rows)
  gemm128_wmma<<<gemmGrid, 128, 0, stream>>>(Qq, WO_w, WO_b, h, h2, M_PAD, N_NODES, 0);
  // 5) hn2 = LN2(h2)  (pad rows zero-filled, never reads h2 pad)
  ln_kernel<<<lnGrid, 256, 0, stream>>>(h2, ln2_g, ln2_b, Kb, N_NODES, M_PAD);
  // 6) t = relu(hn2 @ W1 + b1)
  gemm128_wmma<<<gemmGrid, 128, 0, stream>>>(Kb, W1_w, W1_b, nullptr, Vb, M_PAD, M_PAD, 1);
  // 7) out = h2 + t @ W2 + b2   (store only the N real rows into d_out)
  gemm128_wmma<<<gemmGrid, 128, 0, stream>>>(Vb, W2_w, W2_b, h2, out, M_PAD, N_NODES, 0);

  (void)in_sizes; (void)n_in; (void)out_size; (void)ws_size;
}